// Wavenet_32581621908001
// MI455X (gfx1250) — compile-verified
//
#include <hip/hip_runtime.h>
#include <stdint.h>

typedef __attribute__((ext_vector_type(2))) float v2f;
typedef __attribute__((ext_vector_type(8))) float v8f;

#define NB    8      // batches per block: one per wave32
#define LBUF  1000
#define HID   10

// CDNA5 async global->LDS copy (ASYNCcnt path), GV mode: 64-bit vaddr, saddr=off
__device__ __forceinline__ void async_g2l_b128(unsigned lds_addr, const void* gaddr) {
  asm volatile("global_load_async_to_lds_b128 %0, %1, off"
               :: "v"(lds_addr), "v"(gaddr) : "memory");
}
__device__ __forceinline__ void wait_async0() {
#if __has_builtin(__builtin_amdgcn_s_wait_asynccnt)
  __builtin_amdgcn_s_wait_asynccnt(0);
#else
  asm volatile("s_wait_asynccnt 0" ::: "memory");
#endif
}

// wsh layout: [0..159] w_dil (m*20+h*2+k), [160..239] w_1x1 (m*10+h),
// [240..247] b_1x1, [248..291] w_lin padded to 44 (last 2 zero),
// [292] w_fin, [293] b_fin, [294] b_lin
__global__ __launch_bounds__(256) void wavenet_fused(
    const float* __restrict__ x,     // [B,1,1000]
    const float* __restrict__ wdil,  // [8,10,1,2]
    const float* __restrict__ w1x1,  // [8,1,10,1]
    const float* __restrict__ b1x1,  // [8,1]
    const float* __restrict__ wfin,  // [1,1,1]
    const float* __restrict__ bfin,  // [1]
    const float* __restrict__ wlin,  // [1,42]
    const float* __restrict__ blin,  // [1]
    float* __restrict__ out)         // [B,1]
{
  __shared__ __align__(16) float xbuf[NB][LBUF];  // 32000 B
  __shared__ __align__(16) float accb[NB][44];    // skip accumulators (42 + 2 pad)
  __shared__ float wsh[304];

  const int tid = threadIdx.x;
  const int b0  = blockIdx.x * NB;

  // ---- Phase 1: stage 8 contiguous batch rows (32 KB) via async DMA to LDS ----
  {
    const char* gsrc = (const char*)(x + (size_t)b0 * LBUF);
    const unsigned lbase = (unsigned)(uintptr_t)&xbuf[0][0]; // low32 of flat == LDS offset
    #pragma unroll
    for (int it = 0; it < 8; ++it) {
      const int i = tid + it * 256;               // 16B chunk index, 2000 total
      if (i < (NB * LBUF * 4) / 16)
        async_g2l_b128(lbase + i * 16, gsrc + (size_t)i * 16);
    }
  }
  // tiny weights via plain loads, overlapped with the async DMA
  if (tid < 160)                    wsh[tid] = wdil[tid];
  else if (tid < 240)               wsh[tid] = w1x1[tid - 160];
  else if (tid < 248)               wsh[tid] = b1x1[tid - 240];
  if (tid < 44)                     wsh[248 + tid] = (tid < 42) ? wlin[tid] : 0.0f;
  if (tid == 0) { wsh[292] = wfin[0]; wsh[293] = bfin[0]; wsh[294] = blin[0]; }
  for (int i = tid; i < NB * 44; i += 256) (&accb[0][0])[i] = 0.0f;

  wait_async0();
  __syncthreads();

  // ---- Phase 2: 11-layer recurrence; wave w privately owns batch b0+w in LDS ----
  const int wave = tid >> 5;
  const int lane = tid & 31;

  const int DILS[11] = {2,4,8,16,32,64,64,128,128,256,256};
  const int MODS[11] = {0,1,2,3,4,5,5,6,6,7,7};
  const int LINS[11] = {1000,998,994,986,970,938,874,810,682,554,298};

  #pragma unroll
  for (int li = 0; li < 11; ++li) {
    const int d    = DILS[li];
    const int Lout = LINS[li] - d;
    const int m    = MODS[li];
    const int tail = Lout - 42;

    float w0[HID], w1t[HID], ws[HID];
    #pragma unroll
    for (int c = 0; c < HID; ++c) {
      w0[c]  = wsh[m * 20 + c * 2 + 0];
      w1t[c] = wsh[m * 20 + c * 2 + 1];
      ws[c]  = wsh[160 + m * 10 + c];
    }
    const float bb = wsh[240 + m];

    // In-place update is race-free: every read index in iteration k is strictly
    // greater than every write index of iterations < k (writes at l, reads at
    // l' >= l+1 and l'+d), and same-wave LDS ops are in-order. No barriers.
    for (int l = lane; l < Lout; l += 32) {
      const float x0 = xbuf[wave][l];
      const float x1 = xbuf[wave][l + d];
      float skip = bb;
      #pragma unroll
      for (int c = 0; c < HID; ++c) {
        float y = fmaf(x1, w1t[c], x0 * w0[c]);
        y = fminf(fmaxf(y, -30.0f), 30.0f);      // both factors saturated past here
        // tanh(y)*sigmoid(y) == (1 - e^-y) / (1 + e^-2y):  1 exp + 1 rcp
        const float t = __builtin_amdgcn_exp2f(y * -1.44269504088896340736f);
        const float g = (1.0f - t) * __builtin_amdgcn_rcpf(fmaf(t, t, 1.0f));
        skip = fmaf(g, ws[c], skip);
      }
      xbuf[wave][l] = skip + x1;                 // residual: x_new[l]=skip+x_old[l+d]
      if (l >= tail) accb[wave][l - tail] += skip;  // right-aligned last-42 skip sum
    }
  }

  __syncthreads();

  // ---- Phase 3: out[b] = sum_j (acc[b][j]*wfin + bfin)*wlin[j] + blin ----
  // Batched over the block's 8 batches as 11 chained V_WMMA_F32_16X16X4_F32:
  //   A (16x4): lane<16 -> M=lane, K pair {0,1}; lane>=16 -> M=lane-16, K pair {2,3}
  //   B (4x16): wlin chunk broadcast to all N columns -> every column of D = result
  if (wave == 0) {
    const float vwf = wsh[292], vbf = wsh[293], vbl = wsh[294];
    const int mrow  = lane & 15;
    const int bidx  = mrow & 7;            // clamp pad rows to a valid batch (discarded)
    const int khalf = (lane >> 4) << 1;    // 0 or 2
    v8f c = {0.f, 0.f, 0.f, 0.f, 0.f, 0.f, 0.f, 0.f};
    #pragma unroll
    for (int kc = 0; kc < 11; ++kc) {      // K = 44 (wlin[42..43] = 0)
      const int kb = kc * 4;
      v2f a, bm;
      a.x  = fmaf(accb[bidx][kb + khalf + 0], vwf, vbf);
      a.y  = fmaf(accb[bidx][kb + khalf + 1], vwf, vbf);
      bm.x = wsh[248 + kb + khalf + 0];
      bm.y = wsh[248 + kb + khalf + 1];
      c = __builtin_amdgcn_wmma_f32_16x16x4_f32(false, a, false, bm,
                                                (short)0, c, false, false);
    }
    if (lane == 0) {                       // lane 0 holds D[M=r][N=0] in c[r], r=0..7
      #pragma unroll
      for (int r = 0; r < NB; ++r) out[b0 + r] = c[r] + vbl;
    }
  }
}

extern "C" void kernel_launch(void* const* d_in, const int* in_sizes, int n_in,
                              void* d_out, int out_size, void* d_ws, size_t ws_size,
                              hipStream_t stream) {
  (void)in_sizes; (void)n_in; (void)d_ws; (void)ws_size; (void)out_size;
  const float* x    = (const float*)d_in[0];
  const float* wdil = (const float*)d_in[1];
  const float* w1   = (const float*)d_in[2];
  const float* b1   = (const float*)d_in[3];
  const float* wf   = (const float*)d_in[4];
  const float* bf   = (const float*)d_in[5];
  const float* wl   = (const float*)d_in[6];
  const float* bl   = (const float*)d_in[7];
  float* out = (float*)d_out;
  dim3 grid(32768 / NB), block(256);
  hipLaunchKernelGGL(wavenet_fused, grid, block, 0, stream,
                     x, wdil, w1, b1, wf, bf, wl, bl, out);
}